// customConv2D_4080218931531
// MI455X (gfx1250) — compile-verified
//
#include <hip/hip_runtime.h>

typedef __attribute__((ext_vector_type(2))) float v2f;
typedef __attribute__((ext_vector_type(8))) float v8f;

#define H_IN   256
#define W_IN   256
#define H_OUT  260
#define W_OUT  260
#define TILES_X       17            // ceil(260/16)
#define TILES_PER_IMG (17*17)       // 289
#define WAVES_PER_BLK 8
// total wave-tiles = 512*289 = 147968 = 18496 * 8 exactly
#define N_BLOCKS      18496

__device__ __forceinline__ float pickw(float w0, float w1, float w2,
                                       float w3, float w4, int q) {
    float v = 0.0f;
    v = (q == 0) ? w0 : v;
    v = (q == 1) ? w1 : v;
    v = (q == 2) ? w2 : v;
    v = (q == 3) ? w3 : v;
    v = (q == 4) ? w4 : v;
    return v;
}

__global__ __launch_bounds__(256)
void conv_full_5x5_wmma(const float* __restrict__ in,
                        const float* __restrict__ w,
                        float* __restrict__ out)
{
    const int lane   = threadIdx.x & 31;
    const int waveId = threadIdx.x >> 5;
    const int tile   = blockIdx.x * WAVES_PER_BLK + waveId;

    const int img = tile / TILES_PER_IMG;
    const int rem = tile - img * TILES_PER_IMG;
    const int ty  = rem / TILES_X;
    const int tx  = rem - ty * TILES_X;
    const int y0  = ty * 16;
    const int x0  = tx * 16;

    const int j    = lane & 15;   // A: row M; B: col N; D: col N
    const int half = lane >> 4;   // 0: K=0..1 ; 1: K=2..3
    const int k0   = 2 * half;

    const float* img_in  = in  + (size_t)img * (H_IN * W_IN);
    float*       img_out = out + (size_t)img * (H_OUT * W_OUT);

    // ---- B fragments (Toeplitz band of the 5x5 kernel), hoisted.
    // B_p[c, j] = w[p, j+4-c] for 0 <= j+4-c < 5, else 0 ; c = 4t + k0 + v
    // q masks depend only on (lane, t) -> shared across the 5 weight rows.
    float bx[5][5], by[5][5];   // [p][t]
    #pragma unroll
    for (int p = 0; p < 5; ++p) {
        const float w0 = w[p * 5 + 0];
        const float w1 = w[p * 5 + 1];
        const float w2 = w[p * 5 + 2];
        const float w3 = w[p * 5 + 3];
        const float w4 = w[p * 5 + 4];
        #pragma unroll
        for (int t = 0; t < 5; ++t) {
            const int q0 = j + 4 - (4 * t + k0);
            bx[p][t] = pickw(w0, w1, w2, w3, w4, q0);
            by[p][t] = pickw(w0, w1, w2, w3, w4, q0 - 1);
        }
    }

    v8f acc = {};

    // Wave-uniform interior test: every address this wave touches is in-bounds.
    const bool interior = (ty >= 1) & (ty <= 15) & (tx >= 1) & (tx <= 15);

    if (interior) {
        // ---------- fast path: straight-line, immediate-offset b64 loads ----
        #pragma unroll
        for (int p = 0; p < 5; ++p) {
            const float* rowp = img_in + (size_t)(y0 + j - p) * W_IN + (x0 - 4 + k0);
            #pragma unroll
            for (int t = 0; t < 5; ++t) {
                v2f a = *(const v2f*)(rowp + 4 * t);     // offset:16*t
                v2f b; b.x = bx[p][t]; b.y = by[p][t];
                acc = __builtin_amdgcn_wmma_f32_16x16x4_f32(
                          false, a, false, b, (short)0, acc, false, false);
            }
        }
        #pragma unroll
        for (int rr = 0; rr < 8; ++rr) {
            const int oy = y0 + rr + 8 * half;
            const int ox = x0 + j;
            __builtin_nontemporal_store(acc[rr],
                img_out + (size_t)oy * W_OUT + ox);
        }
    } else {
        // ---------- border path: clamp addresses, mask values (branch-free) --
        #pragma unroll
        for (int p = 0; p < 5; ++p) {
            const int  r     = y0 + j - p;
            const bool rowOK = ((unsigned)r < (unsigned)H_IN);
            const int  rcl   = min(max(r, 0), H_IN - 1);
            const float* rowp = img_in + (size_t)rcl * W_IN;
            #pragma unroll
            for (int t = 0; t < 5; ++t) {
                const int c0  = x0 - 4 + 4 * t + k0;          // even
                const int ccl = min(max(c0, 0), W_IN - 2);    // even -> 8B aligned
                v2f a = *(const v2f*)(rowp + ccl);            // always legal
                a.x = (rowOK && (unsigned)(c0)     < (unsigned)W_IN) ? a.x : 0.0f;
                a.y = (rowOK && (unsigned)(c0 + 1) < (unsigned)W_IN) ? a.y : 0.0f;
                v2f b; b.x = bx[p][t]; b.y = by[p][t];
                acc = __builtin_amdgcn_wmma_f32_16x16x4_f32(
                          false, a, false, b, (short)0, acc, false, false);
            }
        }
        #pragma unroll
        for (int rr = 0; rr < 8; ++rr) {
            const int oy = y0 + rr + 8 * half;
            const int ox = x0 + j;
            if (oy < H_OUT && ox < W_OUT) {
                __builtin_nontemporal_store(acc[rr],
                    img_out + (size_t)oy * W_OUT + ox);
            }
        }
    }
}

extern "C" void kernel_launch(void* const* d_in, const int* in_sizes, int n_in,
                              void* d_out, int out_size, void* d_ws, size_t ws_size,
                              hipStream_t stream) {
    (void)in_sizes; (void)n_in; (void)out_size; (void)d_ws; (void)ws_size;
    const float* x  = (const float*)d_in[0];   // (32,16,256,256) f32
    const float* wt = (const float*)d_in[1];   // (5,5) f32
    float* out = (float*)d_out;                // (32,16,260,260) f32

    conv_full_5x5_wmma<<<dim3(N_BLOCKS), dim3(256), 0, stream>>>(x, wt, out);
}